// MoEFeedForward_28252294873488
// MI455X (gfx1250) — compile-verified
//
#include <hip/hip_runtime.h>
#include <math.h>

typedef __attribute__((ext_vector_type(2))) float v2f;
typedef __attribute__((ext_vector_type(8))) float v8f;
typedef int i32x4 __attribute__((vector_size(16)));   // matches builtin param type

#define T_TOK 65536
#define DIM   256
#define FF    1024
#define NE    8
#define CAP   10240          // ceil(1.25 * 65536 / 8)
#define MB    64
#define MBLKS (CAP / MB)     // 160

// LDS strides (floats). Chosen so the two 16-lane halves of a wave hit
// disjoint bank sets on the 64-bank LDS.
#define ASTR  260            // A tile rows (64 x 256)
#define HSTR  68             // hidden tile rows (64 x 64)
#define W1STR 72             // w1 chunk rows (256 x 64)
#define W2STR 264            // w2 chunk rows (64 x 256)

// dynamic LDS floats: As + Hs + W1s + W2s (separate buffers so async weight
// staging overlaps WMMA compute)
#define SMEM_FLOATS (64 * ASTR + 64 * HSTR + 256 * W1STR + 64 * W2STR)

// ---- gfx1250 async global->LDS path (ASYNCcnt), with sync fallback --------
#if __has_builtin(__builtin_amdgcn_global_load_async_to_lds_b128) && \
    __has_builtin(__builtin_amdgcn_s_wait_asynccnt)
#define HAS_ASYNC 1
#else
#define HAS_ASYNC 0
#endif

#if HAS_ASYNC
__device__ __forceinline__ void async_cp16(const float* g, float* l) {
  // signature (from hipcc diagnostic): (i32x4 AS1*, i32x4 AS3*, Imm offset, Imm cpol)
  __attribute__((address_space(1))) i32x4* gp =
      (__attribute__((address_space(1))) i32x4*)(
          (const __attribute__((address_space(1))) void*)g);
  __attribute__((address_space(3))) i32x4* lp =
      (__attribute__((address_space(3))) i32x4*)(
          (__attribute__((address_space(3))) void*)l);
  __builtin_amdgcn_global_load_async_to_lds_b128(gp, lp, /*offset=*/0, /*cpol=*/0);
}
#endif

__device__ __forceinline__ void wait_async_all() {
#if HAS_ASYNC
  __builtin_amdgcn_s_wait_asynccnt(0);
#endif
}

__device__ __forceinline__ v8f wmma_f32(v2f a, v2f b, v8f c) {
  // D = A(16x4,f32) * B(4x16,f32) + C(16x16,f32)
  return __builtin_amdgcn_wmma_f32_16x16x4_f32(
      /*neg_a=*/false, a, /*neg_b=*/false, b,
      /*c_mod=*/(short)0, c, /*reuse_a=*/false, /*reuse_b=*/false);
}

// ---------------------------------------------------------------- zero out
__global__ void moe_zero_kernel(float4* __restrict__ p, int n4) {
  int i = blockIdx.x * blockDim.x + threadIdx.x;
  int stride = gridDim.x * blockDim.x;
  float4 z = make_float4(0.f, 0.f, 0.f, 0.f);
  for (; i < n4; i += stride) p[i] = z;
}

// ---------------------------------------------------------------- gating
// one wave (32 lanes) per token; 8 tokens per 256-thread block
__global__ __launch_bounds__(256)
void moe_gate_kernel(const float* __restrict__ x,
                     const float* __restrict__ gw,   // [D, E]
                     const float* __restrict__ gb,   // [E]
                     int* __restrict__ idx, float* __restrict__ prob) {
  __shared__ float sgw[DIM * NE];
  __shared__ float sgb[NE];
  for (int i = threadIdx.x; i < DIM * NE; i += 256) sgw[i] = gw[i];
  if (threadIdx.x < NE) sgb[threadIdx.x] = gb[threadIdx.x];
  __syncthreads();

  int lane = threadIdx.x & 31;
  int wid  = threadIdx.x >> 5;
  int t = blockIdx.x * 8 + wid;

  float acc[NE];
#pragma unroll
  for (int e = 0; e < NE; ++e) acc[e] = 0.f;

  const float* xr = x + (size_t)t * DIM;
  for (int d = lane; d < DIM; d += 32) {
    float xv = xr[d];
#pragma unroll
    for (int e = 0; e < NE; ++e) acc[e] += xv * sgw[d * NE + e];
  }
#pragma unroll
  for (int e = 0; e < NE; ++e) {
#pragma unroll
    for (int off = 16; off > 0; off >>= 1)
      acc[e] += __shfl_xor(acc[e], off, 32);
  }
  if (lane == 0) {
    float l[NE];
    float best = -3.4e38f;
    int bi = 0;
#pragma unroll
    for (int e = 0; e < NE; ++e) {
      l[e] = acc[e] + sgb[e];
      if (l[e] > best) { best = l[e]; bi = e; }
    }
    float s = 0.f;
#pragma unroll
    for (int e = 0; e < NE; ++e) s += expf(l[e] - best);
    idx[t] = bi;
    prob[t] = 1.f / s;   // softmax prob of the argmax expert
  }
}

// ---------------------------------------------------------------- exact scan
// one block, wave e handles expert e; ballot-prefix gives the reference's
// ascending-token-order rank exactly and deterministically.
__global__ __launch_bounds__(256)
void moe_scan_kernel(const int* __restrict__ idx,
                     int* __restrict__ tid_list,   // [E, CAP]
                     int* __restrict__ count) {    // [E]
  int lane = threadIdx.x & 31;
  int e    = threadIdx.x >> 5;
  unsigned long long ltmask = (1ull << lane) - 1ull;
  int base = 0;
  for (int t0 = 0; t0 < T_TOK; t0 += 32) {
    int t = t0 + lane;
    int ex = idx[t];
    unsigned long long m = __ballot(ex == e);
    if (ex == e) {
      int p = base + __popcll(m & ltmask);
      if (p < CAP) tid_list[e * CAP + p] = t;
    }
    base += __popcll(m);
  }
  if (lane == 0) count[e] = base;
}

// ---------------------------------------------------------------- fused FFN
// grid: NE * MBLKS workgroups, 256 threads (8 waves).
// Per workgroup: 64 gathered token rows; loop F in chunks of 64:
//   H = relu(A @ w1[:,chunk] + b1)  (WMMA f32 16x16x4)
//   acc += H @ w2[chunk,:]          (WMMA f32 16x16x4, persistent regs)
// Weight chunks are staged into LDS with async-to-LDS loads, double-buffered
// against the WMMA phases.
__global__ __launch_bounds__(256)
void moe_expert_ffn(const float* __restrict__ x,
                    const float* __restrict__ w1,   // [E, D, F]
                    const float* __restrict__ b1,   // [E, F]
                    const float* __restrict__ w2,   // [E, F, D]
                    const float* __restrict__ b2,   // [E, D]
                    const int*  __restrict__ tid_list,
                    const int*  __restrict__ count,
                    const float* __restrict__ prob,
                    float* __restrict__ y) {
  extern __shared__ float sm[];
  float* As  = sm;                       // 64 x ASTR
  float* Hs  = As  + 64 * ASTR;          // 64 x HSTR
  float* W1s = Hs  + 64 * HSTR;          // 256 x W1STR
  float* W2s = W1s + 256 * W1STR;        // 64 x W2STR
  __shared__ int Ts[MB];

  int e    = blockIdx.x / MBLKS;
  int mblk = blockIdx.x % MBLKS;
  int cnt = count[e];
  if (cnt > CAP) cnt = CAP;
  int row0 = mblk * MB;
  if (row0 >= cnt) return;             // uniform early-exit

  int tid = threadIdx.x;
  if (tid < MB) Ts[tid] = (row0 + tid < cnt) ? tid_list[e * CAP + row0 + tid] : -1;
  __syncthreads();

  const float* w1e = w1 + (size_t)e * DIM * FF;
  const float* w2e = w2 + (size_t)e * FF * DIM;
  const float* b1e = b1 + e * FF;
  const float* b2e = b2 + e * DIM;

  // chunk stagers (async issue when available; plain copy otherwise)
  auto stage_w1 = [&](int f0) {
#pragma unroll 4
    for (int i = 0; i < 16; ++i) {
      int fi = i * 256 + tid;            // 4096 x 16B
      int k  = fi >> 4;
      int n4 = fi & 15;
      const float* g = w1e + (size_t)k * FF + f0 + n4 * 4;
      float* l = W1s + k * W1STR + n4 * 4;
#if HAS_ASYNC
      async_cp16(g, l);
#else
      *(float4*)l = *(const float4*)g;
#endif
    }
  };
  auto stage_w2 = [&](int f0) {
#pragma unroll 4
    for (int i = 0; i < 16; ++i) {
      int fi = i * 256 + tid;            // 4096 x 16B
      int f  = fi >> 6;
      int c4 = fi & 63;
      const float* g = w2e + (size_t)(f0 + f) * DIM + c4 * 4;
      float* l = W2s + f * W2STR + c4 * 4;
#if HAS_ASYNC
      async_cp16(g, l);
#else
      *(float4*)l = *(const float4*)g;
#endif
    }
  };

  // gather 64 token rows into LDS (zeros for invalid rows)
#pragma unroll 4
  for (int i = 0; i < 16; ++i) {
    int fi = i * 256 + tid;            // 4096 float4 total
    int r  = fi >> 6;
    int c4 = fi & 63;
    int tok = Ts[r];
    float4 v = make_float4(0.f, 0.f, 0.f, 0.f);
    if (tok >= 0) v = *(const float4*)(x + (size_t)tok * DIM + c4 * 4);
    *(float4*)(As + r * ASTR + c4 * 4) = v;
  }

  // prologue: start fetching first w1 chunk
  stage_w1(0);

  int lane = tid & 31;
  int wid  = tid >> 5;
  int half = lane >> 4;                // 0 or 1
  int l16  = lane & 15;

  // GEMM1 tiling: 4x4 tiles of 16x16; wave -> (mt, {ntp, ntp+1})
  int mt  = wid >> 1;
  int ntp = (wid & 1) * 2;
  // GEMM2 tiling: 4x16 tiles; wave -> (mt2, 8 n-tiles starting ntg*8)
  int mt2 = wid & 3;
  int ntg = wid >> 2;

  v8f accD[8];
#pragma unroll
  for (int n = 0; n < 8; ++n) accD[n] = (v8f){};

  const float* arow  = As + (mt * 16 + l16) * ASTR;
  const float* hrow  = Hs + (mt2 * 16 + l16) * HSTR;
  const float* wcol0 = W1s + ntp * 16 + l16;
  const float* wcol1 = W1s + (ntp + 1) * 16 + l16;
  const float* wcol2 = W2s + l16;
  float* hst0 = Hs + ntp * 16 + l16;
  float* hst1 = Hs + (ntp + 1) * 16 + l16;

  for (int f0 = 0; f0 < FF; f0 += 64) {
    // w1 chunk ready; all waves past previous GEMM2 (W2s free)
    wait_async_all();
    __syncthreads();

    // overlap: fetch w2 chunk while GEMM1 runs
    stage_w2(f0);
    if (f0 + 64 < FF)
      __builtin_prefetch(w1e + (size_t)tid * FF + f0 + 64, 0, 0);

    // ---- GEMM1: H[64, 64] = A[64,256] @ W1chunk[256,64]
    v8f acc0 = (v8f){};
    v8f acc1 = (v8f){};
#pragma unroll 4
    for (int kk = 0; kk < 64; ++kk) {
      int kb = kk * 4 + half * 2;
      v2f a;   a.x   = arow[kb];            a.y   = arow[kb + 1];
      v2f b0;  b0.x  = wcol0[kb * W1STR];   b0.y  = wcol0[(kb + 1) * W1STR];
      v2f b1v; b1v.x = wcol1[kb * W1STR];   b1v.y = wcol1[(kb + 1) * W1STR];
      acc0 = wmma_f32(a, b0, acc0);
      acc1 = wmma_f32(a, b1v, acc1);
    }
    float bias0 = b1e[f0 + ntp * 16 + l16];
    float bias1 = b1e[f0 + (ntp + 1) * 16 + l16];
#pragma unroll
    for (int v = 0; v < 8; ++v) {
      int m = mt * 16 + v + 8 * half;
      hst0[m * HSTR] = fmaxf(acc0[v] + bias0, 0.f);
      hst1[m * HSTR] = fmaxf(acc1[v] + bias1, 0.f);
    }

    // w2 chunk ready; all waves done reading W1s; Hs visible
    wait_async_all();
    __syncthreads();

    // overlap: fetch next w1 chunk while GEMM2 runs
    if (f0 + 64 < FF) stage_w1(f0 + 64);

    // ---- GEMM2: acc[64,256] += H[64,64] @ W2chunk[64,256]
#pragma unroll 2
    for (int kk = 0; kk < 16; ++kk) {
      int kb = kk * 4 + half * 2;
      v2f a; a.x = hrow[kb]; a.y = hrow[kb + 1];
#pragma unroll
      for (int nt = 0; nt < 8; ++nt) {
        int coff = (ntg * 8 + nt) * 16;
        v2f b; b.x = wcol2[kb * W2STR + coff];
               b.y = wcol2[(kb + 1) * W2STR + coff];
        accD[nt] = wmma_f32(a, b, accD[nt]);
      }
    }
  }

  // ---- epilogue: + b2, * gate prob, nan_to_num, scatter to y
  float bias[8];
#pragma unroll
  for (int nt = 0; nt < 8; ++nt) bias[nt] = b2e[(ntg * 8 + nt) * 16 + l16];

#pragma unroll
  for (int v = 0; v < 8; ++v) {
    int m = mt2 * 16 + v + 8 * half;
    int tok = Ts[m];
    if (tok < 0) continue;
    float p = prob[tok];
    float* yr = y + (size_t)tok * DIM + l16;
#pragma unroll
    for (int nt = 0; nt < 8; ++nt) {
      float val = (accD[nt][v] + bias[nt]) * p;
      if (!__builtin_isfinite(val))
        val = (val != val) ? 0.f : ((val > 0.f) ? 1e4f : -1e4f);
      yr[(ntg * 8 + nt) * 16] = val;
    }
  }
}

// ---------------------------------------------------------------- launcher
extern "C" void kernel_launch(void* const* d_in, const int* in_sizes, int n_in,
                              void* d_out, int out_size, void* d_ws, size_t ws_size,
                              hipStream_t stream) {
  (void)in_sizes; (void)n_in; (void)ws_size;
  const float* h  = (const float*)d_in[0];
  const float* gw = (const float*)d_in[1];
  const float* gb = (const float*)d_in[2];
  const float* w1 = (const float*)d_in[3];
  const float* b1 = (const float*)d_in[4];
  const float* w2 = (const float*)d_in[5];
  const float* b2 = (const float*)d_in[6];
  float* y = (float*)d_out;

  // workspace layout
  int*   idx      = (int*)d_ws;                 // T
  float* prob     = (float*)(idx + T_TOK);      // T
  int*   count    = (int*)(prob + T_TOK);       // NE (padded to 64)
  int*   tid_list = count + 64;                 // NE * CAP

  // 1) zero output (dropped tokens -> 0)
  moe_zero_kernel<<<4096, 256, 0, stream>>>((float4*)y, out_size / 4);
  // 2) gating
  moe_gate_kernel<<<T_TOK / 8, 256, 0, stream>>>(h, gw, gb, idx, prob);
  // 3) deterministic per-expert rank / capacity
  moe_scan_kernel<<<1, 256, 0, stream>>>(idx, tid_list, count);
  // 4) fused per-expert FFN with WMMA f32 + async weight staging
  size_t smem = (size_t)SMEM_FLOATS * sizeof(float);
  moe_expert_ffn<<<NE * MBLKS, 256, smem, stream>>>(
      h, w1, b1, w2, b2, tid_list, count, prob, y);
}